// PlanningDecoder_49297634623545
// MI455X (gfx1250) — compile-verified
//
#include <hip/hip_runtime.h>
#include <math.h>

// ---------------------------------------------------------------------------
// Persistent GRU decoder for MI455X (gfx1250).
//  H = OUT = 2048, T = 1024 sequential steps.
//  Weights (167.8 MB fp32) are L2-resident (192 MB L2); all GEMVs via
//  v_wmma_f32_16x16x4_f32, 1 global_load_b128 per WMMA, 4 accumulator chains.
//  One persistent kernel, 5 grid barriers per step.
// ---------------------------------------------------------------------------

#define HDIM     2048
#define TSTEPS   1024
#define NWG      128          // persistent workgroups (all co-resident)
#define NTHREADS 256          // 8 wave32 per WG

typedef __attribute__((ext_vector_type(2))) float v2f;
typedef __attribute__((ext_vector_type(4))) float v4f;
typedef __attribute__((ext_vector_type(8))) float v8f;

__device__ __forceinline__ float sigmoid_f(float x) { return 1.0f / (1.0f + expf(-x)); }

// ---- device-wide generation barrier (persistent kernel sync) --------------
__device__ __forceinline__ void grid_barrier(unsigned* cnt, volatile unsigned* gen) {
  __syncthreads();
  if (threadIdx.x == 0) {
    __builtin_amdgcn_fence(__ATOMIC_RELEASE, "agent");      // flush our stores (global_wb)
    unsigned g = *gen;
    if (atomicAdd(cnt, 1u) == (unsigned)(NWG - 1)) {
      *cnt = 0u;
      __builtin_amdgcn_fence(__ATOMIC_RELEASE, "agent");
      atomicAdd((unsigned*)gen, 1u);                        // release everyone
    } else {
      while (*gen == g) { __builtin_amdgcn_s_sleep(2); }    // low-power spin
    }
    __builtin_amdgcn_fence(__ATOMIC_ACQUIRE, "agent");      // invalidate stale lines (global_inv)
  }
  __syncthreads();
}

// ---- one 16-row GEMV tile: y[row0..row0+15] = W[row0.., :] * x + bias -----
// Whole 256-thread WG cooperates: wave w covers k in [w*256, (w+1)*256).
// K-order inside a WMMA is permuted so each lane loads contiguous 16B:
//   lane<16  (A rows M=lane)   supplies K-slots 0,1 from its quad,
//   lane>=16 (same rows)       supplies K-slots 2,3 from the next quad.
// B[k][n] = x[k] for all n (same per-lane K pattern), so every column of the
// 16x16 D tile equals the GEMV result. Four b128 loads (2 W, 2 x) feed 4 WMMAs on 4
// independent accumulator chains (16 K per iteration).
__device__ __forceinline__ void gemv_tile(const float* __restrict__ W,
                                          const float* __restrict__ x,
                                          const float* __restrict__ bias,
                                          float* __restrict__ y,
                                          int row0, float* sred) {
  const int tid  = threadIdx.x;
  const int lane = tid & 31;
  const int wave = tid >> 5;
  const int mrow = lane & 15;
  const int kq   = (lane >> 4) << 2;                  // 0 for lanes 0-15, 4 for 16-31
  const int k0   = wave * (HDIM / 8) + kq;            // 256-wide K slice per wave

  const float* ap = W + (size_t)(row0 + mrow) * HDIM + k0;
  const float* bp = x + k0;

  v8f acc0 = {}, acc1 = {}, acc2 = {}, acc3 = {};
  #pragma unroll 4
  for (int i = 0; i < (HDIM / 8) / 16; ++i) {         // 16 iterations, 16 K each
    v4f alo = *(const v4f*)(ap);
    v4f blo = *(const v4f*)(bp);
    v4f ahi = *(const v4f*)(ap + 8);
    v4f bhi = *(const v4f*)(bp + 8);
    v2f a0 = __builtin_shufflevector(alo, alo, 0, 1);
    v2f a1 = __builtin_shufflevector(alo, alo, 2, 3);
    v2f a2 = __builtin_shufflevector(ahi, ahi, 0, 1);
    v2f a3 = __builtin_shufflevector(ahi, ahi, 2, 3);
    v2f b0 = __builtin_shufflevector(blo, blo, 0, 1);
    v2f b1 = __builtin_shufflevector(blo, blo, 2, 3);
    v2f b2 = __builtin_shufflevector(bhi, bhi, 0, 1);
    v2f b3 = __builtin_shufflevector(bhi, bhi, 2, 3);
    acc0 = __builtin_amdgcn_wmma_f32_16x16x4_f32(false, a0, false, b0, (short)0, acc0, false, false);
    acc1 = __builtin_amdgcn_wmma_f32_16x16x4_f32(false, a1, false, b1, (short)0, acc1, false, false);
    acc2 = __builtin_amdgcn_wmma_f32_16x16x4_f32(false, a2, false, b2, (short)0, acc2, false, false);
    acc3 = __builtin_amdgcn_wmma_f32_16x16x4_f32(false, a3, false, b3, (short)0, acc3, false, false);
    ap += 16; bp += 16;
  }
  acc0 += acc1;
  acc2 += acc3;
  acc0 += acc2;

  // Extract column 0: lane 0 holds rows 0-7 (VGPR r -> M=r), lane 16 rows 8-15.
  if (lane == 0 || lane == 16) {
    const int rb = (lane >> 4) << 3;
    #pragma unroll
    for (int r = 0; r < 8; ++r) sred[wave * 16 + rb + r] = acc0[r];
  }
  __syncthreads();
  if (tid < 16) {                                     // cross-wave split-K reduction
    float s = 0.0f;
    #pragma unroll
    for (int w = 0; w < 8; ++w) s += sred[w * 16 + tid];
    const int row = row0 + tid;
    y[row] = s + bias[row];
  }
  __syncthreads();
}

// ---------------------------------------------------------------------------
__global__ void __launch_bounds__(NTHREADS, 1)
gru_decoder_init(const float* __restrict__ hidden, float* __restrict__ ws) {
  const int i = blockIdx.x * blockDim.x + threadIdx.x;
  if (i < HDIM) {
    ws[i]        = 0.0f;        // xbuf: GRU input at t=0 is zeros
    ws[HDIM + i] = hidden[i];   // h buffer (parity 0)
  }
  if (i == 0) {
    unsigned* u = (unsigned*)(ws + 13 * HDIM);
    u[0] = 0u;                  // barrier count
    u[1] = 0u;                  // barrier generation
  }
}

__global__ void __launch_bounds__(NTHREADS, 1)
gru_decoder_persistent(const float* __restrict__ w_ih, const float* __restrict__ w_hh,
                       const float* __restrict__ b_ih, const float* __restrict__ b_hh,
                       const float* __restrict__ W1, const float* __restrict__ b1,
                       const float* __restrict__ W2, const float* __restrict__ b2,
                       const float* __restrict__ W3, const float* __restrict__ b3,
                       const float* __restrict__ W4, const float* __restrict__ b4,
                       float* __restrict__ out, float* __restrict__ ws) {
  __shared__ __align__(16) float sm_vec[HDIM];  // h_new (phase 2) / exp buffer (softmax)
  __shared__ float sm_red[256];                 // split-K partials & softmax reductions

  float* xbuf = ws;
  float* hb0  = ws + 1 * HDIM;
  float* hb1  = ws + 2 * HDIM;
  float* gi   = ws + 3 * HDIM;     // 3*HDIM
  float* gh   = ws + 6 * HDIM;     // 3*HDIM
  float* z1   = ws + 9 * HDIM;
  float* z2   = ws + 10 * HDIM;
  float* z3   = ws + 11 * HDIM;
  float* hs   = ws + 12 * HDIM;
  unsigned*          cnt = (unsigned*)(ws + 13 * HDIM);
  volatile unsigned* gen = (volatile unsigned*)(ws + 13 * HDIM) + 1;

  const int wg  = blockIdx.x;
  const int tid = threadIdx.x;

  for (int t = 0; t < TSTEPS; ++t) {
    float*       hcur  = (t & 1) ? hb1 : hb0;
    float*       hnext = (t & 1) ? hb0 : hb1;
    const float* xv    = (t == 0) ? xbuf : hcur;   // x == h for t >= 1

    // ---- phase 1: gate GEMVs. 768 tiles: rows [0,6144)->gi, [6144,12288)->gh
    for (int tl = wg; tl < 768; tl += NWG) {
      const int r0 = tl * 16;
      if (r0 < 3 * HDIM) gemv_tile(w_ih, xv,   b_ih, gi, r0,            sm_red);
      else               gemv_tile(w_hh, hcur, b_hh, gh, r0 - 3 * HDIM, sm_red);
    }
    grid_barrier(cnt, gen);

    // ---- phase 2: h_new (computed redundantly per WG -> LDS, saves a barrier),
    //      wg0 also publishes it for the next step's gates; then z1 = W1*h_new.
    for (int i = tid; i < HDIM; i += NTHREADS) {
      const float ir = gi[i], iz = gi[HDIM + i], inn = gi[2 * HDIM + i];
      const float hr = gh[i], hz = gh[HDIM + i], hn  = gh[2 * HDIM + i];
      const float r  = sigmoid_f(ir + hr);
      const float zg = sigmoid_f(iz + hz);
      const float n  = tanhf(inn + r * hn);
      const float hv = (1.0f - zg) * n + zg * hcur[i];
      sm_vec[i] = hv;
      if (wg == 0) hnext[i] = hv;
    }
    __syncthreads();
    for (int tl = wg; tl < 128; tl += NWG) gemv_tile(W1, sm_vec, b1, z1, tl * 16, sm_red);
    grid_barrier(cnt, gen);

    // ---- phases 3..5: z2, z3, logits
    for (int tl = wg; tl < 128; tl += NWG) gemv_tile(W2, z1, b2, z2, tl * 16, sm_red);
    grid_barrier(cnt, gen);
    for (int tl = wg; tl < 128; tl += NWG) gemv_tile(W3, z2, b3, z3, tl * 16, sm_red);
    grid_barrier(cnt, gen);
    for (int tl = wg; tl < 128; tl += NWG) gemv_tile(W4, z3, b4, hs, tl * 16, sm_red);
    grid_barrier(cnt, gen);

    // ---- phase 6: softmax. Redundant max/sum per WG (reads 8KB from L2),
    //      each WG streams out its own 16-element slice (nontemporal).
    float mx = -3.402823e38f;
    for (int i = tid; i < HDIM; i += NTHREADS) mx = fmaxf(mx, hs[i]);
    sm_red[tid] = mx; __syncthreads();
    for (int s = 128; s > 0; s >>= 1) {
      if (tid < s) sm_red[tid] = fmaxf(sm_red[tid], sm_red[tid + s]);
      __syncthreads();
    }
    const float gmax = sm_red[0]; __syncthreads();

    float ssum = 0.0f;
    for (int i = tid; i < HDIM; i += NTHREADS) {
      const float e = expf(hs[i] - gmax);
      sm_vec[i] = e;
      ssum += e;
    }
    sm_red[tid] = ssum; __syncthreads();
    for (int s = 128; s > 0; s >>= 1) {
      if (tid < s) sm_red[tid] += sm_red[tid + s];
      __syncthreads();
    }
    const float inv = 1.0f / sm_red[0]; __syncthreads();

    if (tid < 16) {
      const int i = wg * 16 + tid;
      __builtin_nontemporal_store(sm_vec[i] * inv, out + (size_t)t * HDIM + i);
    }
    // No barrier needed here: next phase 1 touches only gi/gh/h buffers,
    // all protected by the barriers above.
  }
}

// ---------------------------------------------------------------------------
extern "C" void kernel_launch(void* const* d_in, const int* in_sizes, int n_in,
                              void* d_out, int out_size, void* d_ws, size_t ws_size,
                              hipStream_t stream) {
  (void)in_sizes; (void)n_in; (void)out_size; (void)ws_size;
  const float* hidden = (const float*)d_in[0];
  // d_in[1] = target: only defines T (fixed at 1024); values unused by reference.
  const float* w_ih = (const float*)d_in[2];
  const float* w_hh = (const float*)d_in[3];
  const float* b_ih = (const float*)d_in[4];
  const float* b_hh = (const float*)d_in[5];
  const float* W1   = (const float*)d_in[6];
  const float* b1   = (const float*)d_in[7];
  const float* W2   = (const float*)d_in[8];
  const float* b2   = (const float*)d_in[9];
  const float* W3   = (const float*)d_in[10];
  const float* b3   = (const float*)d_in[11];
  const float* W4   = (const float*)d_in[12];
  const float* b4   = (const float*)d_in[13];
  float* out = (float*)d_out;
  float* ws  = (float*)d_ws;

  hipLaunchKernelGGL(gru_decoder_init, dim3((HDIM + NTHREADS - 1) / NTHREADS),
                     dim3(NTHREADS), 0, stream, hidden, ws);
  hipLaunchKernelGGL(gru_decoder_persistent, dim3(NWG), dim3(NTHREADS), 0, stream,
                     w_ih, w_hh, b_ih, b_hh, W1, b1, W2, b2, W3, b3, W4, b4, out, ws);
}